// GPUKmeans_8246337209000
// MI455X (gfx1250) — compile-verified
//
#include <hip/hip_runtime.h>
#include <stdint.h>

// ---------------------------------------------------------------------------
// Batched K-means for MI455X (gfx1250).  fp32 WMMA (V_WMMA_F32_16X16X4_F32)
// drives both GEMMs; X/C tiles are staged into LDS with gfx1250 async
// global->LDS copies (ASYNCcnt) so the WMMA fragment loads hit LDS.
// ---------------------------------------------------------------------------

#define BATCH 4
#define NPTS  8192
#define DIM   64
#define KC    32
#define MAXIT 30
#define TIE_EPS 1e-8f

typedef float v2f __attribute__((ext_vector_type(2)));
typedef float v8f __attribute__((ext_vector_type(8)));

#if defined(__gfx1250__) && __has_builtin(__builtin_amdgcn_global_load_async_to_lds_b128)
#define HAVE_ASYNC_LDS 1
#endif

#if defined(HAVE_ASYNC_LDS)
typedef int v4i_vs __attribute__((vector_size(16)));
typedef __attribute__((address_space(1))) v4i_vs* async_gptr;
typedef __attribute__((address_space(3))) v4i_vs* async_lptr;
#endif

__device__ __forceinline__ void async_copy_b128(const float* g, float* l) {
#if defined(HAVE_ASYNC_LDS)
  __builtin_amdgcn_global_load_async_to_lds_b128((async_gptr)g, (async_lptr)l,
                                                 0, 0);
#else
  float4 t = *(const float4*)g;
  *(float4*)l = t;
#endif
}

__device__ __forceinline__ void async_wait_all() {
#if defined(HAVE_ASYNC_LDS)
#if __has_builtin(__builtin_amdgcn_s_wait_asynccnt)
  __builtin_amdgcn_s_wait_asynccnt(0);
#else
  asm volatile("s_wait_asynccnt 0x0" ::: "memory");
#endif
#endif
}

__device__ __forceinline__ v8f wmma_f32x4(v2f a, v2f b, v8f c) {
  // D = A(16x4 f32) * B(4x16 f32) + C(16x16 f32), wave32
  return __builtin_amdgcn_wmma_f32_16x16x4_f32(false, a, false, b, (short)0, c,
                                               false, false);
}

// Order-preserving float <-> uint mapping for atomic min/max on any-sign floats
__device__ __forceinline__ uint32_t enc_f(float f) {
  uint32_t u = __float_as_uint(f);
  return (u & 0x80000000u) ? ~u : (u | 0x80000000u);
}
__device__ __forceinline__ float dec_f(uint32_t u) {
  return __uint_as_float((u & 0x80000000u) ? (u ^ 0x80000000u) : ~u);
}

// ---------------------------------------------------------------------------
// Distance tile: block = 256 threads (8 waves).  Stages the 128-point X tile
// (32 KB) and the full centroid matrix C (8 KB) into LDS via async copies,
// then computes Dm[p][k] = ||x_p||^2 + ||c_k||^2 - 2 x_p.c_k for 128 x 32
// into LDS, with fp32 WMMA for the x.c inner products (fragments from LDS).
// ---------------------------------------------------------------------------
__device__ __forceinline__ void compute_dist_tile(
    const float* __restrict__ Xb, const float* __restrict__ Cb,
    float* lds_X /*128*DIM*/, float* lds_C /*KC*DIM*/, float* lds_cc /*KC*/,
    float* lds_xx /*128*/, float* lds_dm /*128*KC*/, int pblock) {
  const int tid  = threadIdx.x;
  const int wave = tid >> 5;
  const int lane = tid & 31;

  // ---- async stage: X tile (8 x B128 per thread-slot) + C (2 x B128) ----
  {
    const float* gx = Xb + (size_t)pblock * DIM;
#pragma unroll
    for (int j = 0; j < (128 * DIM) / (256 * 4); ++j) {  // 8
      const int off = (j * 256 + tid) * 4;
      async_copy_b128(gx + off, lds_X + off);
    }
#pragma unroll
    for (int j = 0; j < (KC * DIM) / (256 * 4); ++j) {   // 2
      const int off = (j * 256 + tid) * 4;
      async_copy_b128(Cb + off, lds_C + off);
    }
    async_wait_all();
  }
  __syncthreads();

  // ---- norms from LDS ----
  if (tid < KC) {
    const float* cp = lds_C + tid * DIM;
    float s = 0.f;
#pragma unroll 8
    for (int d = 0; d < DIM; ++d) s += cp[d] * cp[d];
    lds_cc[tid] = s;
  }
  if (tid < 128) {
    const float* xp = lds_X + tid * DIM;
    float s = 0.f;
#pragma unroll 8
    for (int d = 0; d < DIM; ++d) s += xp[d] * xp[d];
    lds_xx[tid] = s;
  }
  __syncthreads();

  const int row   = lane & 15;           // A: M row / B: N col
  const int koff  = (lane >> 4) << 1;    // K sub-offset 0 or 2 per lane-half
  const int half8 = (lane >> 4) << 3;    // C/D row offset per lane-half

  const float* xp  = lds_X + (wave * 16 + row) * DIM;
  const float* cp0 = lds_C + row * DIM;          // clusters 0..15
  const float* cp1 = lds_C + (row + 16) * DIM;   // clusters 16..31

  v8f acc0 = {0.f, 0.f, 0.f, 0.f, 0.f, 0.f, 0.f, 0.f};
  v8f acc1 = acc0;
#pragma unroll
  for (int kk = 0; kk < DIM; kk += 4) {
    v2f a, b0, b1;
    a.x  = xp[kk + koff];   a.y  = xp[kk + koff + 1];
    b0.x = cp0[kk + koff];  b0.y = cp0[kk + koff + 1];
    b1.x = cp1[kk + koff];  b1.y = cp1[kk + koff + 1];
    acc0 = wmma_f32x4(a, b0, acc0);
    acc1 = wmma_f32x4(a, b1, acc1);
  }
#pragma unroll
  for (int v = 0; v < 8; ++v) {
    const int m   = v + half8;            // point row within wave tile
    const float x = lds_xx[wave * 16 + m];
    lds_dm[(wave * 16 + m) * KC + row]      = x + lds_cc[row]      - 2.f * acc0[v];
    lds_dm[(wave * 16 + m) * KC + row + 16] = x + lds_cc[row + 16] - 2.f * acc1[v];
  }
  __syncthreads();
}

#define DIST_TILE_SHARED                                  \
  __shared__ __align__(16) float lds_X[128 * DIM];        \
  __shared__ __align__(16) float lds_C[KC * DIM];         \
  __shared__ float lds_cc[KC];                            \
  __shared__ float lds_xx[128];                           \
  __shared__ float lds_dm[128 * KC];

// ---------------------------------------------------------------------------
// Init / clear kernels
// ---------------------------------------------------------------------------
__global__ void k_initX(const float* __restrict__ E, float* __restrict__ X,
                        uint32_t* __restrict__ mask, uint32_t* __restrict__ flags) {
  size_t i = (size_t)blockIdx.x * blockDim.x + threadIdx.x;
  if (i < (size_t)BATCH * NPTS * DIM) X[i] = E[i];
  if (i < (size_t)BATCH * NPTS) mask[i] = 0u;
  if (i == 0) { flags[0] = 0u; flags[1] = 0u; }
}

__global__ void k_initC(const float* __restrict__ E, float* __restrict__ C) {
  int i = blockIdx.x * blockDim.x + threadIdx.x;  // B*KC*DIM
  if (i >= BATCH * KC * DIM) return;
  int d = i & (DIM - 1);
  int k = (i >> 6) & (KC - 1);
  int b = i >> 11;
  uint32_t h = 0x9E3779B9u * (uint32_t)(k + 1);   // deterministic pseudo-perm
  h ^= h >> 16; h *= 0x85EBCA6Bu; h ^= h >> 13; h *= 0xC2B2AE35u; h ^= h >> 16;
  int sel = (int)(h & (NPTS - 1));
  C[i] = E[((size_t)b * NPTS + sel) * DIM + d];
}

__global__ void k_clear_iter(float* Cacc, uint32_t* maxw, uint32_t* cnt,
                             float* fat, uint32_t* diffs, uint32_t* flags) {
  int i = blockIdx.x * blockDim.x + threadIdx.x;
  if (i < BATCH * KC * DIM) Cacc[i] = 0.f;
  if (i < BATCH * KC) { maxw[i] = 0u; cnt[i] = 0u; fat[i] = 0.f; }
  if (i < BATCH) diffs[i] = 0u;
  if (i == 0) flags[1] = 0u;
}

// ---------------------------------------------------------------------------
// Assignment pass: WMMA distances -> per-point argmin/tie mask, diff count,
// per-cluster log-W max (for stable softmax) and member counts.
// ---------------------------------------------------------------------------
__global__ void __launch_bounds__(256)
k_assign(const float* __restrict__ X, const float* __restrict__ C,
         const float* __restrict__ logW, const uint32_t* __restrict__ mask,
         uint32_t* __restrict__ newmask, uint32_t* __restrict__ maxw,
         uint32_t* __restrict__ cnt, uint32_t* __restrict__ diffs,
         const uint32_t* __restrict__ flags) {
  if (flags[0]) return;  // converged: state frozen
  DIST_TILE_SHARED
  const int b = blockIdx.y;
  const int pblock = blockIdx.x * 128;
  compute_dist_tile(X + (size_t)b * NPTS * DIM, C + (size_t)b * KC * DIM,
                    lds_X, lds_C, lds_cc, lds_xx, lds_dm, pblock);
  const int tid = threadIdx.x;
  if (tid < 128) {
    const float* row = &lds_dm[tid * KC];
    float mind = row[0];
#pragma unroll
    for (int k = 1; k < KC; ++k) mind = fminf(mind, row[k]);
    uint32_t m = 0u;
#pragma unroll
    for (int k = 0; k < KC; ++k)
      if (fabsf(row[k] - mind) < TIE_EPS) m |= (1u << k);
    const size_t gi = (size_t)b * NPTS + pblock + tid;
    newmask[gi] = m;
    const uint32_t changed = (uint32_t)__builtin_popcount(m ^ mask[gi]);
    if (changed) atomicAdd(&diffs[b], changed);
    const uint32_t e = enc_f(logW[gi]);
    uint32_t mm = m;
    while (mm) {
      const int k = __builtin_ctz(mm);
      mm &= mm - 1;
      atomicMax(&maxw[b * KC + k], e);
      atomicAdd(&cnt[b * KC + k], 1u);
    }
  }
}

// Convergence + deterministic replacement-index generation
__global__ void k_done(uint32_t* flags, const uint32_t* diffs, uint32_t* repl, int it) {
  if (threadIdx.x == 0) {
    bool allzero = true;
    for (int b = 0; b < BATCH; ++b)
      if (diffs[b] != 0u) allzero = false;   // diff/N < 1e-7  <=>  diff == 0
    const uint32_t done = flags[0] | (allzero ? 1u : 0u);
    flags[0] = done;
    flags[1] = done ? 0u : 1u;               // commit this iteration?
  }
  const int i = threadIdx.x;                 // b*KC + k
  if (i < BATCH * KC) {
    uint32_t h = (uint32_t)it * 2654435761u ^ (uint32_t)(i + 1) * 0x9E3779B9u;
    h ^= h >> 16; h *= 0x85EBCA6Bu; h ^= h >> 13; h *= 0xC2B2AE35u; h ^= h >> 16;
    repl[i] = h & (NPTS - 1);
  }
}

__global__ void k_commit(uint32_t* __restrict__ mask,
                         const uint32_t* __restrict__ newmask,
                         const uint32_t* __restrict__ flags) {
  if (!flags[1]) return;
  const int i = blockIdx.x * blockDim.x + threadIdx.x;
  if (i < BATCH * NPTS) mask[i] = newmask[i];
}

// Empty-cluster replacement: gather (old X) then apply, two kernels to avoid
// read/write aliasing among the first K rows.
__global__ void k_gather(const float* __restrict__ X, float* __restrict__ stage,
                         const uint32_t* __restrict__ repl,
                         const uint32_t* __restrict__ cnt,
                         const uint32_t* __restrict__ flags) {
  if (!flags[1]) return;
  const int i = blockIdx.x * blockDim.x + threadIdx.x;  // B*KC*DIM
  if (i >= BATCH * KC * DIM) return;
  const int d = i & (DIM - 1), k = (i >> 6) & (KC - 1), b = i >> 11;
  if (cnt[b * KC + k] != 0u) return;
  stage[i] = X[((size_t)b * NPTS + repl[b * KC + k]) * DIM + d];
}

__global__ void k_apply(float* __restrict__ X, const float* __restrict__ stage,
                        const uint32_t* __restrict__ cnt,
                        const uint32_t* __restrict__ flags) {
  if (!flags[1]) return;
  const int i = blockIdx.x * blockDim.x + threadIdx.x;
  if (i >= BATCH * KC * DIM) return;
  const int d = i & (DIM - 1), k = (i >> 6) & (KC - 1), b = i >> 11;
  if (cnt[b * KC + k] != 0u) return;
  X[((size_t)b * NPTS + k) * DIM + d] = stage[i];
}

// Softmax denominator per cluster: fat[b,k] = sum_n w[n,k]
__global__ void k_fat(const uint32_t* __restrict__ mask,
                      const float* __restrict__ logW,
                      const uint32_t* __restrict__ maxw, float* __restrict__ fat,
                      const uint32_t* __restrict__ flags) {
  if (!flags[1]) return;
  const int i = blockIdx.x * blockDim.x + threadIdx.x;  // B*N
  if (i >= BATCH * NPTS) return;
  const int b = i >> 13;
  uint32_t m = mask[i];
  const float lw = logW[i];
  while (m) {
    const int k = __builtin_ctz(m);
    m &= m - 1;
    atomicAdd(&fat[b * KC + k], __expf(lw - dec_f(maxw[b * KC + k])));
  }
}

// ---------------------------------------------------------------------------
// Centroid update GEMM via fp32 WMMA: Cacc[k,d] += sum_n w[n,k] * X[n,d].
// A = w^T (16 clusters x 4 points), B = X (4 points x 16 dims).
// Block = 256 threads / 8 waves, each wave reduces its own 32-point span.
// ---------------------------------------------------------------------------
__global__ void __launch_bounds__(256)
k_centroid(const float* __restrict__ X, const float* __restrict__ logW,
           const uint32_t* __restrict__ mask, const uint32_t* __restrict__ maxw,
           float* __restrict__ Cacc, const uint32_t* __restrict__ flags) {
  if (!flags[1]) return;
  __shared__ float wlds[256][KC];     // softmax weights for this point chunk
  const int b = blockIdx.y;
  const int chunk = blockIdx.x * 256;
  const int tid = threadIdx.x;
  {
    const size_t gi = (size_t)b * NPTS + chunk + tid;
    const uint32_t m = mask[gi];
    const float lw = logW[gi];
#pragma unroll
    for (int k = 0; k < KC; ++k)
      wlds[tid][k] =
          ((m >> k) & 1u) ? __expf(lw - dec_f(maxw[b * KC + k])) : 0.f;
  }
  __syncthreads();

  const int wave = tid >> 5, lane = tid & 31;
  const int row   = lane & 15;
  const int koff  = (lane >> 4) << 1;
  const int half8 = (lane >> 4) << 3;
  const int wbase = wave * 32;        // this wave's 32-point span in the chunk
  const float* Xb = X + ((size_t)b * NPTS + chunk + wbase) * DIM;

  const v8f z = {0.f, 0.f, 0.f, 0.f, 0.f, 0.f, 0.f, 0.f};
  v8f acc[2][4];
#pragma unroll
  for (int kt = 0; kt < 2; ++kt)
#pragma unroll
    for (int dt = 0; dt < 4; ++dt) acc[kt][dt] = z;

#pragma unroll
  for (int p = 0; p < 32; p += 4) {
    const int lp = wbase + p + koff;
    v2f a0, a1;
    a0.x = wlds[lp][row];      a0.y = wlds[lp + 1][row];       // clusters 0..15
    a1.x = wlds[lp][row + 16]; a1.y = wlds[lp + 1][row + 16];  // clusters 16..31
    const float* x0 = Xb + (size_t)(p + koff) * DIM;
#pragma unroll
    for (int dt = 0; dt < 4; ++dt) {
      v2f bb;
      bb.x = x0[dt * 16 + row];
      bb.y = x0[DIM + dt * 16 + row];
      acc[0][dt] = wmma_f32x4(a0, bb, acc[0][dt]);
      acc[1][dt] = wmma_f32x4(a1, bb, acc[1][dt]);
    }
  }
  float* Cab = Cacc + (size_t)b * KC * DIM;
#pragma unroll
  for (int kt = 0; kt < 2; ++kt)
#pragma unroll
    for (int dt = 0; dt < 4; ++dt)
#pragma unroll
      for (int v = 0; v < 8; ++v) {
        const int c = kt * 16 + v + half8;
        const int d = dt * 16 + row;
        atomicAdd(&Cab[c * DIM + d], acc[kt][dt][v]);
      }
}

__global__ void k_finishC(float* __restrict__ C, const float* __restrict__ Cacc,
                          const float* __restrict__ fat,
                          const uint32_t* __restrict__ flags) {
  if (!flags[1]) return;
  const int i = blockIdx.x * blockDim.x + threadIdx.x;  // B*KC*DIM
  if (i >= BATCH * KC * DIM) return;
  C[i] = Cacc[i] / fmaxf(fat[i >> 6], 1e-12f);
}

// ---------------------------------------------------------------------------
// Final outputs
// ---------------------------------------------------------------------------
__global__ void k_clear_final(float* repAcc, float* obj, uint32_t* rmin,
                              uint32_t* ridx, uint32_t* rcnt) {
  const int i = blockIdx.x * blockDim.x + threadIdx.x;
  if (i < BATCH * KC * DIM) repAcc[i] = 0.f;
  if (i < BATCH * KC) { rmin[i] = 0xFFFFFFFFu; ridx[i] = 0xFFFFFFFFu; rcnt[i] = 0u; }
  if (i < BATCH) obj[i] = 0.f;
}

__global__ void __launch_bounds__(256)
k_final1(const float* __restrict__ X, const float* __restrict__ C,
         float* __restrict__ obj, uint32_t* __restrict__ rmin) {
  DIST_TILE_SHARED
  const int b = blockIdx.y;
  const int pblock = blockIdx.x * 128;
  compute_dist_tile(X + (size_t)b * NPTS * DIM, C + (size_t)b * KC * DIM,
                    lds_X, lds_C, lds_cc, lds_xx, lds_dm, pblock);
  const int tid = threadIdx.x;
  if (tid < 128) {
    const float* row = &lds_dm[tid * KC];
    float mind = row[0];
#pragma unroll
    for (int k = 1; k < KC; ++k) mind = fminf(mind, row[k]);
    atomicAdd(&obj[b], mind * (1.f / NPTS));
#pragma unroll
    for (int k = 0; k < KC; ++k) atomicMin(&rmin[b * KC + k], enc_f(row[k]));
  }
}

__global__ void __launch_bounds__(256)
k_final2(const float* __restrict__ X, const float* __restrict__ C,
         const uint32_t* __restrict__ rmin, uint32_t* __restrict__ ridx,
         uint32_t* __restrict__ rcnt, float* __restrict__ repAcc) {
  DIST_TILE_SHARED
  const int b = blockIdx.y;
  const int pblock = blockIdx.x * 128;
  compute_dist_tile(X + (size_t)b * NPTS * DIM, C + (size_t)b * KC * DIM,
                    lds_X, lds_C, lds_cc, lds_xx, lds_dm, pblock);
  const int tid = threadIdx.x;
  if (tid < 128) {
    const int n = pblock + tid;
    const float* xp = lds_X + tid * DIM;     // this point's row, already in LDS
    const float* row = &lds_dm[tid * KC];
    for (int k = 0; k < KC; ++k) {
      const uint32_t mn = rmin[b * KC + k];
      const float dval = row[k];
      if (enc_f(dval) == mn) atomicMin(&ridx[b * KC + k], (uint32_t)n);
      if (fabsf(dval - dec_f(mn)) < TIE_EPS) {
        atomicAdd(&rcnt[b * KC + k], 1u);
        for (int d = 0; d < DIM; ++d)
          atomicAdd(&repAcc[(b * KC + k) * DIM + d], xp[d]);
      }
    }
  }
}

__global__ void k_output(float* __restrict__ out, const uint32_t* __restrict__ mask,
                         const float* __restrict__ C, const float* __restrict__ repAcc,
                         const uint32_t* __restrict__ rcnt,
                         const uint32_t* __restrict__ ridx,
                         const float* __restrict__ obj) {
  const size_t i = (size_t)blockIdx.x * blockDim.x + threadIdx.x;
  const size_t nAss = (size_t)BATCH * NPTS * KC;          // 2^20
  if (i < nAss) {
    const int k = (int)(i & 31);
    out[i] = ((mask[i >> 5] >> k) & 1u) ? 1.f : 0.f;
  }
  if (i < (size_t)BATCH * KC * DIM) {
    out[nAss + i] = C[i];
    const float c = fmaxf((float)rcnt[i >> 6], 0.001f);
    out[nAss + BATCH * KC * DIM + i] = repAcc[i] / c;
  }
  if (i < (size_t)BATCH * KC)
    out[nAss + 2 * BATCH * KC * DIM + i] = (float)ridx[i];
  if (i < (size_t)BATCH)
    out[nAss + 2 * BATCH * KC * DIM + BATCH * KC + i] = obj[i];
}

// ---------------------------------------------------------------------------
// Host side
// ---------------------------------------------------------------------------
extern "C" void kernel_launch(void* const* d_in, const int* in_sizes, int n_in,
                              void* d_out, int out_size, void* d_ws,
                              size_t ws_size, hipStream_t stream) {
  (void)in_sizes; (void)n_in; (void)out_size; (void)ws_size;
  const float* E    = (const float*)d_in[0];
  const float* logW = (const float*)d_in[1];
  float* out = (float*)d_out;

  constexpr size_t NXD = (size_t)BATCH * NPTS * DIM;  // 2,097,152
  constexpr size_t NKD = (size_t)BATCH * KC * DIM;    // 8,192
  constexpr size_t NBK = (size_t)BATCH * KC;          // 128
  constexpr size_t NBN = (size_t)BATCH * NPTS;        // 32,768

  float* ws      = (float*)d_ws;
  float* X       = ws;
  float* C       = X + NXD;
  float* Cacc    = C + NKD;
  float* stage   = Cacc + NKD;
  float* repAcc  = stage + NKD;
  float* fat     = repAcc + NKD;
  float* obj     = fat + NBK;
  uint32_t* u    = (uint32_t*)(obj + BATCH);
  uint32_t* mask    = u;
  uint32_t* newmask = mask + NBN;
  uint32_t* maxw    = newmask + NBN;
  uint32_t* cnt     = maxw + NBK;
  uint32_t* diffs   = cnt + NBK;
  uint32_t* repl    = diffs + BATCH;
  uint32_t* rmin    = repl + NBK;
  uint32_t* ridx    = rmin + NBK;
  uint32_t* rcnt    = ridx + NBK;
  uint32_t* flags   = rcnt + NBK;

  // Re-initialize all state each launch (determinism across graph replays).
  k_initX<<<(unsigned)(NXD / 256), 256, 0, stream>>>(E, X, mask, flags);
  k_initC<<<(unsigned)(NKD / 256), 256, 0, stream>>>(E, C);

  for (int it = 0; it < MAXIT; ++it) {
    k_clear_iter<<<(unsigned)(NKD / 256), 256, 0, stream>>>(Cacc, maxw, cnt,
                                                            fat, diffs, flags);
    k_assign<<<dim3(NPTS / 128, BATCH), 256, 0, stream>>>(
        X, C, logW, mask, newmask, maxw, cnt, diffs, flags);
    k_done<<<1, 128, 0, stream>>>(flags, diffs, repl, it);
    k_commit<<<(unsigned)(NBN / 256), 256, 0, stream>>>(mask, newmask, flags);
    k_gather<<<(unsigned)(NKD / 256), 256, 0, stream>>>(X, stage, repl, cnt, flags);
    k_apply<<<(unsigned)(NKD / 256), 256, 0, stream>>>(X, stage, cnt, flags);
    k_fat<<<(unsigned)(NBN / 256), 256, 0, stream>>>(mask, logW, maxw, fat, flags);
    k_centroid<<<dim3(NPTS / 256, BATCH), 256, 0, stream>>>(X, logW, mask, maxw,
                                                            Cacc, flags);
    k_finishC<<<(unsigned)(NKD / 256), 256, 0, stream>>>(C, Cacc, fat, flags);
  }

  k_clear_final<<<(unsigned)(NKD / 256), 256, 0, stream>>>(repAcc, obj, rmin,
                                                           ridx, rcnt);
  k_final1<<<dim3(NPTS / 128, BATCH), 256, 0, stream>>>(X, C, obj, rmin);
  k_final2<<<dim3(NPTS / 128, BATCH), 256, 0, stream>>>(X, C, rmin, ridx, rcnt,
                                                        repAcc);
  k_output<<<(unsigned)(((size_t)BATCH * NPTS * KC) / 256), 256, 0, stream>>>(
      out, mask, C, repAcc, rcnt, ridx, obj);
}